// DecoderA_20813411516876
// MI455X (gfx1250) — compile-verified
//
#include <hip/hip_runtime.h>
#include <math.h>

#define B_SZ   128
#define M_CHK  288
#define N_VAR  576
#define K_INFO 288
#define N_ITER 5
#define NW     18        // N_VAR / 32 = dwords per packed H row

typedef __attribute__((ext_vector_type(2))) float v2f;
typedef __attribute__((ext_vector_type(8))) float v8f;

// ---------------------------------------------------------------------------
// Kernel 0: pack binary H [B,M,N] f32 -> bitmask (18 dwords per row) via
// wave32 ballot. Shrinks per-iteration H traffic 32x and drops the L2
// working set from ~170MB to ~88MB (fully resident in the 192MB L2).
// ---------------------------------------------------------------------------
__global__ __launch_bounds__(256) void pack_kernel(
    const float* __restrict__ H, unsigned* __restrict__ Hbits) {
  const int wave = threadIdx.x >> 5;
  const int lane = threadIdx.x & 31;
  const int row = blockIdx.x * 8 + wave;        // row = b*M_CHK + m
  const float* hrow = H + (size_t)row * N_VAR;
  unsigned myword = 0;
#pragma unroll
  for (int j = 0; j < NW; ++j) {
    const float hv = hrow[j * 32 + lane];
    const unsigned msk = (unsigned)__ballot(hv > 0.0f);   // wave32: low 32 bits
    if (lane == j) myword = msk;
  }
  if (lane < NW) Hbits[(size_t)row * NW + lane] = myword;
}

// ---------------------------------------------------------------------------
// Kernel 1: norm_llrs[b,n] = llr / mean(|llr|) with llr = -4*inputs/sigma2.
// Also zero-init the pooled accumulator (ws is not re-poisoned between calls).
// ---------------------------------------------------------------------------
__global__ __launch_bounds__(288) void setup_kernel(
    const float* __restrict__ inputs, const float* __restrict__ sigma2,
    float* __restrict__ norm_llrs, float* __restrict__ pooled) {
  __shared__ float red[288];
  const int b = blockIdx.x;
  const int tid = threadIdx.x;
  const float inv_s2 = -4.0f / sigma2[b];
  const int n0 = tid, n1 = tid + 288;
  const float l0 = inputs[(size_t)b * N_VAR + n0] * inv_s2;
  const float l1 = inputs[(size_t)b * N_VAR + n1] * inv_s2;
  red[tid] = fabsf(l0) + fabsf(l1);
  __syncthreads();
  for (int s = 144; s > 8; s >>= 1) {           // 144,72,36,18,9
    if (tid < s) red[tid] += red[tid + s];
    __syncthreads();
  }
  float tot = 0.0f;
#pragma unroll
  for (int i = 0; i < 9; ++i) tot += red[i];
  const float inv_mean = (float)N_VAR / tot;
  norm_llrs[(size_t)b * N_VAR + n0] = l0 * inv_mean;
  norm_llrs[(size_t)b * N_VAR + n1] = l1 * inv_mean;
  pooled[(size_t)b * N_VAR + n0] = 0.0f;
  pooled[(size_t)b * N_VAR + n1] = 0.0f;
}

// ---------------------------------------------------------------------------
// Kernel 2: column sums via WMMA (S = ones(16x4) x M-tiles, accumulated over
// m in steps of 4), then fused posterior / normalization / pooling for the
// *previous* iteration; on the last call also emits sigmoid(-out).
// Block = 288 threads = 9 full wave32s (EXEC all ones for WMMA).
// ---------------------------------------------------------------------------
__global__ __launch_bounds__(288) void colsum_kernel(
    const float* __restrict__ Mst, const float* __restrict__ norm_llrs,
    float* __restrict__ Sg, float* __restrict__ pooled,
    const float* __restrict__ ip, const float* __restrict__ op,
    const float* __restrict__ skip, float* __restrict__ d_out,
    int t_prev, int final_flag) {
  __shared__ float Slds[N_VAR];
  __shared__ float red[288];
  const int b = blockIdx.x;
  const int tid = threadIdx.x;
  const int wave = tid >> 5;
  const int lane = tid & 31;
  const int colInTile = lane & 15;
  const int rowSel = (lane >> 4) << 1;           // lanes 0-15 -> K rows 0,1; 16-31 -> 2,3
  const float* Mb = Mst + (size_t)b * M_CHK * N_VAR;

  v2f a; a.x = 1.0f; a.y = 1.0f;                 // all-ones A matrix (16x4)
#pragma unroll
  for (int i = 0; i < 4; ++i) {                  // 9 waves * 4 tiles = 36 tiles = 576 cols
    const int tile = wave * 4 + i;
    const int col = tile * 16 + colInTile;
    v8f c = {0.f, 0.f, 0.f, 0.f, 0.f, 0.f, 0.f, 0.f};
    const float* base = Mb + (size_t)rowSel * N_VAR + col;
    for (int m0 = 0; m0 < M_CHK; m0 += 4) {
      v2f bb;
      bb.x = base[0];                            // row m0+rowSel
      bb.y = base[N_VAR];                        // row m0+rowSel+1
      __builtin_prefetch(base + 8 * N_VAR, 0, 3);   // near-cache prefetch
      c = __builtin_amdgcn_wmma_f32_16x16x4_f32(false, a, false, bb,
                                                (short)0, c, false, false);
      base += 4 * N_VAR;
    }
    const float sv = c[0];                       // D[0,col] = colsum
    if (lane < 16) {
      Slds[col] = sv;
      Sg[(size_t)b * N_VAR + col] = sv;
    }
  }
  __syncthreads();

  // posterior of iteration t_prev, per-row abs-mean normalization, pooling
  const float ipt = ip[t_prev];
  const float opt = op[t_prev];
  const int n0 = tid, n1 = tid + 288;
  const float x0 = norm_llrs[(size_t)b * N_VAR + n0];
  const float x1 = norm_llrs[(size_t)b * N_VAR + n1];
  const float p0 = x0 * ipt + Slds[n0];
  const float p1 = x1 * ipt + Slds[n1];
  red[tid] = fabsf(p0) + fabsf(p1);
  __syncthreads();
  for (int s = 144; s > 8; s >>= 1) {
    if (tid < s) red[tid] += red[tid + s];
    __syncthreads();
  }
  float tot = 0.0f;
#pragma unroll
  for (int i = 0; i < 9; ++i) tot += red[i];
  const float inv_mean = (float)N_VAR / tot;

  const float pl0 = pooled[(size_t)b * N_VAR + n0] + opt * p0 * inv_mean;
  const float pl1 = pooled[(size_t)b * N_VAR + n1] + opt * p1 * inv_mean;
  pooled[(size_t)b * N_VAR + n0] = pl0;
  pooled[(size_t)b * N_VAR + n1] = pl1;

  if (final_flag) {                              // n0 = tid < 288 == K_INFO
    const float outv = pl0 * (1.0f / (float)N_ITER) + skip[0] * x0;
    d_out[(size_t)b * K_INFO + n0] = 1.0f / (1.0f + expf(outv));
  }
}

// ---------------------------------------------------------------------------
// Kernel 3: check-node row update. One wave32 per (b,m) row; each lane owns
// 18 strided columns. Edge set comes from the packed bitmask; M loads/stores
// and all transcendentals are EXEC-predicated on edges (non-edge state is
// provably 0 after iteration 0, which performs the full overwrite).
// ---------------------------------------------------------------------------
__global__ __launch_bounds__(256) void row_kernel(
    const unsigned* __restrict__ Hbits, const float* __restrict__ norm_llrs,
    const float* __restrict__ Sg, const float* __restrict__ w_cv,
    const float* __restrict__ ip, const float* __restrict__ gate_logit,
    float* __restrict__ Mst, int t, int first) {
  const int wave = threadIdx.x >> 5;
  const int lane = threadIdx.x & 31;
  const int row = blockIdx.x * 8 + wave;         // row = b*M_CHK + m
  const int b = row / M_CHK;
  const int m = row - b * M_CHK;

  const float ipt = ip[t];
  const float gate = 1.0f / (1.0f + expf(-gate_logit[0]));
  const float omg = 1.0f - gate;

  const unsigned* hb = Hbits + (size_t)row * NW;
  const float* wrow = w_cv + (size_t)m * N_VAR;
  const float* xrow = norm_llrs + (size_t)b * N_VAR;
  const float* srow = Sg + (size_t)b * N_VAR;
  float* mrow = Mst + (size_t)row * N_VAR;

  unsigned emask = 0;                            // bit j: lane's element j is an edge
  float tv[NW], mold[NW];
  float p = 1.0f;
#pragma unroll
  for (int j = 0; j < NW; ++j) {
    const unsigned hw = hb[j];                   // wave-uniform broadcast load
    const bool edge = (hw >> lane) & 1u;
    emask |= (edge ? 1u : 0u) << j;
    const int n = lane + 32 * j;
    float mo = 0.0f;
    float tval = 1.0f;                           // non-edge contributes 1 to product
    if (edge) {
      mo = first ? 0.0f : mrow[n];
      const float x = xrow[n] * ipt;
      const float S = first ? 0.0f : srow[n];
      float V = x + S - mo;
      V = fminf(fmaxf(V, -15.0f), 15.0f);
      tval = tanhf(0.5f * V);
    }
    mold[j] = mo;
    tv[j] = tval;
    p *= tval;
  }
  // full-wave product reduction (wave32)
#pragma unroll
  for (int off = 16; off > 0; off >>= 1) p *= __shfl_xor(p, off, 32);

#pragma unroll
  for (int j = 0; j < NW; ++j) {
    const int n = lane + 32 * j;
    const bool edge = (emask >> j) & 1u;
    float outv = 0.0f;
    if (edge) {
      const float tval = tv[j];
      const float ts = (fabsf(tval) < 1e-7f)
                           ? ((tval >= 0.0f) ? 1e-7f : -1e-7f) : tval;
      float r = p / ts;
      r = fminf(fmaxf(r, -1.0f + 1e-6f), 1.0f - 1e-6f);
      const float mnew = logf((1.0f + r) / (1.0f - r)) * wrow[n];  // 2*atanh(r)*w
      outv = gate * mnew + omg * mold[j];
    }
    // iteration 0: full overwrite (d_ws poisoned); later: edges only
    if (first || edge) mrow[n] = outv;
  }
}

// ---------------------------------------------------------------------------
extern "C" void kernel_launch(void* const* d_in, const int* in_sizes, int n_in,
                              void* d_out, int out_size, void* d_ws, size_t ws_size,
                              hipStream_t stream) {
  const float* inputs     = (const float*)d_in[0];
  const float* H          = (const float*)d_in[1];
  const float* sigma2     = (const float*)d_in[2];
  const float* ip         = (const float*)d_in[3];
  const float* op         = (const float*)d_in[4];
  const float* skip       = (const float*)d_in[5];
  const float* w_cv       = (const float*)d_in[6];
  const float* gate_logit = (const float*)d_in[7];
  float* out = (float*)d_out;

  float* ws = (float*)d_ws;
  const size_t BMN = (size_t)B_SZ * M_CHK * N_VAR;
  const size_t BN  = (size_t)B_SZ * N_VAR;
  float* Mst        = ws;            // 85 MB message state (L2-resident)
  float* norm_llrs  = ws + BMN;
  float* Sg         = norm_llrs + BN;
  float* pooled     = Sg + BN;
  unsigned* Hbits   = (unsigned*)(pooled + BN);  // 2.65 MB packed mask

  pack_kernel<<<(B_SZ * M_CHK) / 8, 256, 0, stream>>>(H, Hbits);
  setup_kernel<<<B_SZ, 288, 0, stream>>>(inputs, sigma2, norm_llrs, pooled);

  for (int t = 0; t < N_ITER; ++t) {
    if (t > 0)
      colsum_kernel<<<B_SZ, 288, 0, stream>>>(Mst, norm_llrs, Sg, pooled,
                                              ip, op, skip, out, t - 1, 0);
    row_kernel<<<(B_SZ * M_CHK) / 8, 256, 0, stream>>>(
        Hbits, norm_llrs, Sg, w_cv, ip, gate_logit, Mst, t, (t == 0) ? 1 : 0);
  }
  // final colsum: posterior of iteration 4 + pooling + sigmoid output
  colsum_kernel<<<B_SZ, 288, 0, stream>>>(Mst, norm_llrs, Sg, pooled,
                                          ip, op, skip, out, N_ITER - 1, 1);
}